// HEALDownSampler_40518721470591
// MI455X (gfx1250) — compile-verified
//
#include <hip/hip_runtime.h>
#include <math.h>

typedef float v2f __attribute__((ext_vector_type(2)));
typedef float v8f __attribute__((ext_vector_type(8)));

#define NB    2
#define NE    196608
#define NR    49152
#define FIN   128
#define LOUT  128
#define MTOT  (NB * NR)     // 98304 output rows
#define MBLK  64            // rows per workgroup
#define XS_ST 132           // padded LDS row stride (floats) -> conflict-free A frags

__device__ __forceinline__ float gelu_tanh(float v) {
    // matches jax.nn.gelu(approximate=True)
    const float k0 = 0.7978845608028654f;   // sqrt(2/pi)
    const float k1 = 0.044715f;
    float u = k0 * (v + k1 * v * v * v);
    return 0.5f * v * (1.0f + tanhf(u));
}

__global__ __launch_bounds__(256, 1) void heal_fused(
    const float* __restrict__ x,
    const float* __restrict__ edge_attr,
    const float* __restrict__ we1, const float* __restrict__ be1,
    const float* __restrict__ we2, const float* __restrict__ be2,
    const float* __restrict__ wf1, const float* __restrict__ bf1,
    const float* __restrict__ wf2, const float* __restrict__ bf2,
    float* __restrict__ out)
{
    __shared__ __align__(16) float lds_xs[MBLK * XS_ST];  // 33,792 B (also scratch for stage 0)
    __shared__ float lds_c1[LOUT];

    const int tid  = threadIdx.x;
    const int wave = tid >> 5;
    const int lane = tid & 31;
    const int half = lane >> 4;        // 0: lanes 0-15, 1: lanes 16-31
    const int l16  = lane & 15;
    const int ncol = wave * 16 + l16;  // this wave's output column (0..127)

    // ---------------- Stage 0: fold the edge-embedding MLP into c1[128] ----------------
    float* g    = lds_xs;        // 256 gelu(attr_j * we1 + be1) values
    float* gs   = lds_xs + 256;  // 64: sum over j
    float* esum = lds_xs + 320;  // 64: aggregated embedding (constant over all pixels)
    {
        int j = tid >> 6, i = tid & 63;
        float a = edge_attr[j];
        g[tid] = gelu_tanh(a * we1[i] + be1[i]);
    }
    __syncthreads();
    if (tid < 64) gs[tid] = g[tid] + g[64 + tid] + g[128 + tid] + g[192 + tid];
    __syncthreads();
    if (tid < 64) {
        float s = 4.0f * be2[tid];
        for (int i = 0; i < 64; ++i) s += gs[i] * we2[i * 64 + tid];
        esum[tid] = s;
    }
    __syncthreads();
    if (tid < 128) {
        float s = bf1[tid];
        for (int c = 0; c < 64; ++c) s += esum[c] * wf1[c * LOUT + tid];
        lds_c1[tid] = s;
    }

    // ---------------- Persistent B fragments in VGPRs (wave owns 16 columns) ----------------
    // B-matrix 4x16 layout: V0 = {K=0 | K=2}, V1 = {K=1 | K=3} across half-waves.
    v2f b1[32], b2[32];
    const float* W1 = wf1 + 64 * LOUT;   // rows 64..191 multiply the x-part of the concat
#pragma unroll
    for (int kc = 0; kc < 32; ++kc) {
        int r0 = 4 * kc + 2 * half;
        b1[kc].x = W1[(size_t)r0 * LOUT + ncol];
        b1[kc].y = W1[(size_t)(r0 + 1) * LOUT + ncol];
        b2[kc].x = wf2[(size_t)r0 * LOUT + ncol];
        b2[kc].y = wf2[(size_t)(r0 + 1) * LOUT + ncol];
    }
    __syncthreads();   // c1 done; lds_xs scratch free for reuse

    // ---------------- xs = sum of 4 child rows, staged into LDS ----------------
    const int gm0 = blockIdx.x * MBLK;
#pragma unroll
    for (int k = 0; k < 8; ++k) {
        int p   = tid + k * 256;      // float4 slot: 64 rows * 32 slots
        int row = p >> 5;
        int c4  = p & 31;
        int gm  = gm0 + row;
        int b   = gm / NR;
        int r   = gm - b * NR;
        const float4* src = (const float4*)(x + ((size_t)b * NE + (size_t)4 * r) * FIN) + c4;
        float4 s0 = src[0];
        float4 s1 = src[32];          // +1 child row (128 floats)
        float4 s2 = src[64];
        float4 s3 = src[96];
        float4 a4;
        a4.x = s0.x + s1.x + s2.x + s3.x;
        a4.y = s0.y + s1.y + s2.y + s3.y;
        a4.z = s0.z + s1.z + s2.z + s3.z;
        a4.w = s0.w + s1.w + s2.w + s3.w;
        *(float4*)&lds_xs[row * XS_ST + c4 * 4] = a4;
    }
    __syncthreads();

    // ---------------- GEMM1: h = gelu(xs @ W1 + c1), kept in registers ----------------
    const float c1n = lds_c1[ncol];
    v8f h[4];
#pragma unroll
    for (int s = 0; s < 4; ++s) {
        v8f acc = {};
        const float* arow = &lds_xs[(s * 16 + l16) * XS_ST + 2 * half];
#pragma unroll
        for (int kc = 0; kc < 32; ++kc) {
            v2f a = *(const v2f*)&arow[4 * kc];   // A frag: {K=4kc+2h, K=4kc+2h+1}
            acc = __builtin_amdgcn_wmma_f32_16x16x4_f32(
                false, a, false, b1[kc], (short)0, acc, false, false);
        }
#pragma unroll
        for (int v = 0; v < 8; ++v) h[s][v] = gelu_tanh(acc[v] + c1n);
    }
    __syncthreads();               // everyone done reading xs
#pragma unroll
    for (int s = 0; s < 4; ++s) {
#pragma unroll
        for (int v = 0; v < 8; ++v) {
            int row = s * 16 + v + 8 * half;      // C/D layout: M = v (+8 for high half)
            lds_xs[row * XS_ST + ncol] = h[s][v];
        }
    }
    __syncthreads();

    // ---------------- GEMM2: out = h @ wf2 + bf2 ----------------
    const float b2n = bf2[ncol];
#pragma unroll
    for (int s = 0; s < 4; ++s) {
        v8f acc = {};
        const float* arow = &lds_xs[(s * 16 + l16) * XS_ST + 2 * half];
#pragma unroll
        for (int kc = 0; kc < 32; ++kc) {
            v2f a = *(const v2f*)&arow[4 * kc];
            acc = __builtin_amdgcn_wmma_f32_16x16x4_f32(
                false, a, false, b2[kc], (short)0, acc, false, false);
        }
#pragma unroll
        for (int v = 0; v < 8; ++v) {
            int gm = gm0 + s * 16 + v + 8 * half;
            out[(size_t)gm * LOUT + ncol] = acc[v] + b2n;
        }
    }
}

extern "C" void kernel_launch(void* const* d_in, const int* in_sizes, int n_in,
                              void* d_out, int out_size, void* d_ws, size_t ws_size,
                              hipStream_t stream) {
    const float* x   = (const float*)d_in[0];
    const float* ea  = (const float*)d_in[1];
    // d_in[2] = edge_rec (int32): structure (r = edge/4) folded into indexing
    const float* we1 = (const float*)d_in[3];
    const float* be1 = (const float*)d_in[4];
    const float* we2 = (const float*)d_in[5];
    const float* be2 = (const float*)d_in[6];
    const float* wf1 = (const float*)d_in[7];
    const float* bf1 = (const float*)d_in[8];
    const float* wf2 = (const float*)d_in[9];
    const float* bf2 = (const float*)d_in[10];
    float* out = (float*)d_out;

    dim3 grid(MTOT / MBLK);   // 1536 workgroups
    dim3 block(256);          // 8 waves (wave32)
    hipLaunchKernelGGL(heal_fused, grid, block, 0, stream,
                       x, ea, we1, be1, we2, be2, wf1, bf1, wf2, bf2, out);
}